// Filtration_82222853914919
// MI455X (gfx1250) — compile-verified
//
#include <hip/hip_runtime.h>
#include <math.h>

typedef __attribute__((ext_vector_type(2))) float v2f;
typedef __attribute__((ext_vector_type(8))) float v8f;

#define GDIM 64
#define SLOPE 0.01f
#define BN_EPS 1e-5f
#define CHUNK 256

// Build x0 = concat(emb_deg[deg], emb_lab[lab]) directly into Z[:, 0:128], ldZ=320
__global__ void k_embed(const float* __restrict__ emb_deg,
                        const float* __restrict__ emb_lab,
                        const int* __restrict__ ndeg,
                        const int* __restrict__ nlab,
                        float* __restrict__ Z, int total) {
  int tid = blockIdx.x * blockDim.x + threadIdx.x;
  if (tid >= total) return;
  int i = tid >> 5, q = tid & 31;
  float4 v;
  if (q < 16) v = ((const float4*)(emb_deg + (size_t)ndeg[i] * GDIM))[q];
  else        v = ((const float4*)(emb_lab + (size_t)nlab[i] * GDIM))[q - 16];
  ((float4*)(Z + (size_t)i * 320))[q] = v;
}

// H[i, 0:D] = X[i*ldX, 0:D]  (float4 granularity)
__global__ void k_copy(const float* __restrict__ X, int ldX,
                       float* __restrict__ H, int D, int total) {
  int tid = blockIdx.x * blockDim.x + threadIdx.x;
  if (tid >= total) return;
  int vec = D >> 2;
  int i = tid / vec, d4 = (tid % vec) << 2;
  *(float4*)(H + (size_t)i * D + d4) = *(const float4*)(X + (size_t)i * ldX + d4);
}

// H[dst] += X[src]  over all edges (segment_sum)
__global__ void k_scatter(const float* __restrict__ X, int ldX,
                          float* __restrict__ H, int D,
                          const int* __restrict__ src, const int* __restrict__ dst,
                          int total) {
  int tid = blockIdx.x * blockDim.x + threadIdx.x;
  if (tid >= total) return;
  int vec = D >> 2;
  int e = tid / vec, d4 = (tid % vec) << 2;
  int s = src[e], t = dst[e];
  float4 v = *(const float4*)(X + (size_t)s * ldX + d4);
  float* hp = H + (size_t)t * D + d4;
  atomicAdd(hp + 0, v.x);
  atomicAdd(hp + 1, v.y);
  atomicAdd(hp + 2, v.z);
  atomicAdd(hp + 3, v.w);
}

// Y[N, DOUT] = act(X[N,K] @ W[DOUT,K]^T + bias), DOUT = 16 * (blockDim.x/32).
// Each wave computes a 32x16 output slab (two 16x16 WMMA tiles sharing one B
// fragment per k-step -> 2x register reuse of weights, two independent
// accumulation chains). fp32 WMMA 16x16x4: A lane=M, vgpr=K pair, upper
// half-wave K+2; B mirrored; C/D standard 8-VGPR tile.
__global__ void k_gemm_wmma(const float* __restrict__ X, int ldX,
                            const float* __restrict__ W,
                            const float* __restrict__ bias,
                            float* __restrict__ Y, int ldY,
                            int N, int K, int act) {
  extern __shared__ float smem[];              // 32 x (K+4) A panel (padded)
  const int KP = K + 4;
  int lane = threadIdx.x & 31;
  int wave = threadIdx.x >> 5;
  int row0 = blockIdx.x << 5;
  int col0 = wave << 4;

  // cooperative vectorized A-panel load; clamp tail rows so waves stay full
  int kv = K >> 2;
  for (int idx = threadIdx.x; idx < 32 * kv; idx += blockDim.x) {
    int r = idx / kv, c4 = (idx % kv) << 2;
    int rr = row0 + r; if (rr >= N) rr = N - 1;
    *(float4*)(smem + r * KP + c4) = *(const float4*)(X + (size_t)rr * ldX + c4);
  }
  __syncthreads();

  int m = lane & 15;
  int hi = lane >> 4;                          // 0: K+0/1, 1: K+2/3
  int wcol = col0 + m;                         // B/N index for this lane
  const float* wrow = W + (size_t)wcol * K;
  __builtin_prefetch(wrow, 0, 1);              // global_prefetch_b8

  v8f acc0 = {}, acc1 = {};
  for (int k0 = 0; k0 < K; k0 += 4) {
    int ka = k0 + 2 * hi;
    v2f a0, a1, b;
    b.x  = wrow[ka];
    b.y  = wrow[ka + 1];
    a0.x = smem[m * KP + ka];
    a0.y = smem[m * KP + ka + 1];
    a1.x = smem[(m + 16) * KP + ka];
    a1.y = smem[(m + 16) * KP + ka + 1];
    acc0 = __builtin_amdgcn_wmma_f32_16x16x4_f32(false, a0, false, b,
                                                 (short)0, acc0, false, false);
    acc1 = __builtin_amdgcn_wmma_f32_16x16x4_f32(false, a1, false, b,
                                                 (short)0, acc1, false, false);
  }

  float bs = bias[wcol];
#pragma unroll
  for (int r = 0; r < 8; r++) {
    int row = row0 + r + 8 * hi;               // C/D layout: M = r (+8 upper half)
    if (row < N) {
      float v = acc0[r] + bs;
      if (act) v = v > 0.f ? v : SLOPE * v;
      Y[(size_t)row * ldY + wcol] = v;
    }
    int row2 = row + 16;
    if (row2 < N) {
      float v = acc1[r] + bs;
      if (act) v = v > 0.f ? v : SLOPE * v;
      Y[(size_t)row2 * ldY + wcol] = v;
    }
  }
}

__global__ void k_zero(float* __restrict__ p, int n) {
  int tid = blockIdx.x * blockDim.x + threadIdx.x;
  if (tid < n) p[tid] = 0.f;
}

// Column sums / sums of squares over N rows -> ST[0:D], ST[D:2D]
__global__ void k_colstats(const float* __restrict__ X, int ldX, int D, int N,
                           float* __restrict__ ST) {
  __shared__ float ps[256];
  int r0 = blockIdx.x * CHUNK;
  int rows = N - r0; if (rows > CHUNK) rows = CHUNK;
  for (int j = threadIdx.x; j < 2 * D; j += blockDim.x) ps[j] = 0.f;
  __syncthreads();
  float s = 0.f, sq = 0.f;
  int col = threadIdx.x % D;                   // constant per thread (256 % D == 0)
  for (int idx = threadIdx.x; idx < rows * D; idx += blockDim.x) {
    float x = X[(size_t)(r0 + idx / D) * ldX + (idx % D)];
    s += x; sq += x * x;
  }
  atomicAdd(&ps[col], s);
  atomicAdd(&ps[D + col], sq);
  __syncthreads();
  for (int j = threadIdx.x; j < 2 * D; j += blockDim.x) atomicAdd(&ST[j], ps[j]);
}

// y = act((x - mean) * rsqrt(var+eps) * gw + gb)
__global__ void k_bn_apply(const float* __restrict__ X, int ldX,
                           float* __restrict__ Y, int ldY,
                           const float* __restrict__ gw, const float* __restrict__ gb,
                           const float* __restrict__ ST,
                           int D, int N, float invN, int act) {
  int tid = blockIdx.x * blockDim.x + threadIdx.x;
  if (tid >= N * D) return;
  int i = tid / D, j = tid % D;
  float mean = ST[j] * invN;
  float var  = ST[D + j] * invN - mean * mean;
  float x = X[(size_t)i * ldX + j];
  float v = (x - mean) * rsqrtf(var + BN_EPS) * gw[j] + gb[j];
  if (act) v = v > 0.f ? v : SLOPE * v;
  Y[(size_t)i * ldY + j] = v;
}

// out[i] = sigmoid(dot(H[i,:64], w) + b); one wave32 per node
__global__ void k_fc2(const float* __restrict__ H, const float* __restrict__ w,
                      const float* __restrict__ b, float* __restrict__ out, int N) {
  int i = blockIdx.x * (blockDim.x >> 5) + (threadIdx.x >> 5);
  int lane = threadIdx.x & 31;
  if (i >= N) return;
  const float* h = H + (size_t)i * GDIM;
  float v = h[lane] * w[lane] + h[lane + 32] * w[lane + 32];
  for (int off = 16; off; off >>= 1) v += __shfl_xor(v, off, 32);
  if (lane == 0) out[i] = 1.f / (1.f + expf(-(v + b[0])));
}

extern "C" void kernel_launch(void* const* d_in, const int* in_sizes, int n_in,
                              void* d_out, int out_size, void* d_ws, size_t ws_size,
                              hipStream_t stream) {
  const float* emb_deg = (const float*)d_in[0];
  const float* emb_lab = (const float*)d_in[1];
  const float* wa[3] = {(const float*)d_in[2],  (const float*)d_in[8],  (const float*)d_in[14]};
  const float* ba[3] = {(const float*)d_in[3],  (const float*)d_in[9],  (const float*)d_in[15]};
  const float* wb[3] = {(const float*)d_in[4],  (const float*)d_in[10], (const float*)d_in[16]};
  const float* bb[3] = {(const float*)d_in[5],  (const float*)d_in[11], (const float*)d_in[17]};
  const float* gw[3] = {(const float*)d_in[6],  (const float*)d_in[12], (const float*)d_in[18]};
  const float* gb[3] = {(const float*)d_in[7],  (const float*)d_in[13], (const float*)d_in[19]};
  const float* fc1_w  = (const float*)d_in[20];
  const float* fc1_b  = (const float*)d_in[21];
  const float* fcbn_w = (const float*)d_in[22];
  const float* fcbn_b = (const float*)d_in[23];
  const float* fc2_w  = (const float*)d_in[24];
  const float* fc2_b  = (const float*)d_in[25];
  const int* ndeg = (const int*)d_in[26];
  const int* nlab = (const int*)d_in[27];
  const int* eidx = (const int*)d_in[28];

  int N = in_sizes[26];
  int E = in_sizes[28] / 2;
  const int* src = eidx;
  const int* dst = eidx + E;

  float* Z  = (float*)d_ws;                 // [N, 320] concat buffer
  float* H  = Z  + (size_t)N * 320;         // [N, 128] agg / final-BN scratch
  float* T1 = H  + (size_t)N * 128;         // [N, 128] mlp hidden
  float* T2 = T1 + (size_t)N * 128;         // [N, 64]  mlp out
  float* ST = T2 + (size_t)N * 64;          // [256]    BN stats
  float invN = 1.f / (float)N;

  dim3 b256(256);
  int slabTiles = (N + 31) / 32;            // 32 output rows per block

  k_embed<<<(N * 32 + 255) / 256, b256, 0, stream>>>(emb_deg, emb_lab, ndeg, nlab, Z, N * 32);

  for (int l = 0; l < 3; l++) {
    int Din = (l == 0) ? 128 : 64;
    const float* X = (l == 0) ? Z : (Z + 128 + 64 * (l - 1));
    int ct = N * (Din / 4);
    k_copy<<<(ct + 255) / 256, b256, 0, stream>>>(X, 320, H, Din, ct);
    int et = E * (Din / 4);
    k_scatter<<<(et + 255) / 256, b256, 0, stream>>>(X, 320, H, Din, src, dst, et);

    size_t lds = (size_t)32 * (Din + 4) * sizeof(float);
    k_gemm_wmma<<<slabTiles, 32 * (Din / 16), lds, stream>>>(H, Din, wa[l], ba[l], T1, Din, N, Din, 1);
    k_gemm_wmma<<<slabTiles, 32 * (64 / 16),  lds, stream>>>(T1, Din, wb[l], bb[l], T2, 64, N, Din, 0);

    k_zero<<<1, 128, 0, stream>>>(ST, 128);
    k_colstats<<<(N + CHUNK - 1) / CHUNK, b256, 0, stream>>>(T2, 64, 64, N, ST);
    k_bn_apply<<<(N * 64 + 255) / 256, b256, 0, stream>>>(T2, 64, Z + 128 + 64 * l, 320,
                                                          gw[l], gb[l], ST, 64, N, invN, 1);
  }

  size_t lds2 = (size_t)32 * (320 + 4) * sizeof(float);
  k_gemm_wmma<<<slabTiles, 128, lds2, stream>>>(Z, 320, fc1_w, fc1_b, T1, 64, N, 320, 0);
  k_zero<<<1, 128, 0, stream>>>(ST, 128);
  k_colstats<<<(N + CHUNK - 1) / CHUNK, b256, 0, stream>>>(T1, 64, 64, N, ST);
  k_bn_apply<<<(N * 64 + 255) / 256, b256, 0, stream>>>(T1, 64, H, 64,
                                                        fcbn_w, fcbn_b, ST, 64, N, invN, 1);
  k_fc2<<<(N + 7) / 8, b256, 0, stream>>>(H, fc2_w, fc2_b, (float*)d_out, N);
}